// Titans_24489903522014
// MI455X (gfx1250) — compile-verified
//
#include <hip/hip_runtime.h>
#include <math.h>

// ---------------------------------------------------------------------------
// Titans (memory-update + cross-attention + FC/ReLU) for gfx1250 (CDNA5).
// All GEMMs use V_WMMA_F32_16X16X4_F32 (exact fp32 matrix path, wave32).
// Shapes: B=2 S=1024 D=128 M=512 H=512 heads=4 hd=32, N=B*S=2048.
// ---------------------------------------------------------------------------

typedef __attribute__((ext_vector_type(2))) float v2f;
typedef __attribute__((ext_vector_type(8))) float v8f;

#define NROW 2048   // B*S
#define MMEM 512
#define DDIM 128
#define HOUT 512

// A-frag / B-frag(for X@W^T) loader: row-major matrix, returns the two f32
// this lane owns per the ISA 16x4 layout (lane<16: k0,k0+1 ; lane>=16: k0+2,k0+3).
__device__ __forceinline__ v2f frag_rm(const float* __restrict__ p, int ld,
                                       int rbase, int cbase) {
  int lane = threadIdx.x & 31;
  const float* q = p + (size_t)(rbase + (lane & 15)) * ld +
                   (cbase + ((lane >> 4) << 1));
  float2 t = *(const float2*)q;               // 8B aligned: even col, even ld
  v2f r; r.x = t.x; r.y = t.y;
  return r;
}

// Operand accessed "down a column": element (kbase+k, nbase+n) of a row-major
// array. Used for B of X@Y (Y row-major over k) and A of sur^T (sur[k][m]).
__device__ __forceinline__ v2f frag_cm(const float* __restrict__ p, int ld,
                                       int kbase, int nbase) {
  int lane = threadIdx.x & 31;
  int k = kbase + ((lane >> 4) << 1);
  int n = nbase + (lane & 15);
  v2f r;
  r.x = p[(size_t)k * ld + n];
  r.y = p[(size_t)(k + 1) * ld + n];
  return r;
}

__device__ __forceinline__ v8f wmma4(v2f a, v2f b, v8f c) {
  // D = A(16x4) * B(4x16) + C, fp32 exact
  return __builtin_amdgcn_wmma_f32_16x16x4_f32(false, a, false, b,
                                               (short)0, c, false, false);
}

// --------------------------- row norms -------------------------------------
// one wave per row; rows [0,2048)=x, rows [2048,2560)=memory
__global__ void norms_kernel(const float* __restrict__ x,
                             const float* __restrict__ mem,
                             float* __restrict__ xn, float* __restrict__ mn) {
  int row = blockIdx.x * (blockDim.x >> 5) + (threadIdx.x >> 5);
  int lane = threadIdx.x & 31;
  const float* src;
  float* dst;
  if (row < NROW) { src = x + (size_t)row * DDIM; dst = xn + row; }
  else            { src = mem + (size_t)(row - NROW) * DDIM; dst = mn + (row - NROW); }
  float s = 0.f;
#pragma unroll
  for (int i = 0; i < 4; ++i) { float v = src[lane + 32 * i]; s += v * v; }
#pragma unroll
  for (int m = 16; m >= 1; m >>= 1) s += __shfl_xor(s, m, 32);
  if (lane == 0) *dst = s;
}

// ------------------- G = x @ mem^T  ->  surprise ---------------------------
// 4096 tiles of 16x16, one per wave, K=128 (32 wmma steps)
__global__ void surprise_kernel(const float* __restrict__ x,
                                const float* __restrict__ mem,
                                const float* __restrict__ xn,
                                const float* __restrict__ mn,
                                float* __restrict__ sur) {
  int wave = blockIdx.x * (blockDim.x >> 5) + (threadIdx.x >> 5);
  int rbase = (wave >> 5) * 16;          // 128 row tiles
  int cbase = (wave & 31) * 16;          // 32 col tiles
  v8f acc = {};
#pragma unroll
  for (int k = 0; k < DDIM; k += 4) {
    v2f a = frag_rm(x, DDIM, rbase, k);
    v2f b = frag_rm(mem, DDIM, cbase, k);   // B[k][m] = mem[m][k]
    acc = wmma4(a, b, acc);
  }
  int lane = threadIdx.x & 31;
  int cOff = lane & 15, rHi = (lane >> 4) * 8;
#pragma unroll
  for (int i = 0; i < 8; ++i) {
    int r = rbase + rHi + i, c = cbase + cOff;
    float s2 = xn[r] - 2.f * acc[i] + mn[c];
    sur[(size_t)r * MMEM + c] = sqrtf(fmaxf(s2, 0.f));
  }
}

// ------------------------ s1[m] = colsum(sur) ------------------------------
__global__ void colsum_kernel(const float* __restrict__ sur,
                              float* __restrict__ s1) {
  __shared__ float red[256];
  int m = blockIdx.x;
  float s = 0.f;
  for (int n = threadIdx.x; n < NROW; n += 256) s += sur[(size_t)n * MMEM + m];
  red[threadIdx.x] = s;
  __syncthreads();
  for (int off = 128; off >= 1; off >>= 1) {
    if ((int)threadIdx.x < off) red[threadIdx.x] += red[threadIdx.x + off];
    __syncthreads();
  }
  if (threadIdx.x == 0) s1[m] = red[0];
}

// ----- mem_new = mem + (LR/N)*(sur^T @ x - s1 * mem) -----------------------
// 256 tiles (32 m-tiles x 8 d-tiles), K=2048
__global__ void memupdate_kernel(const float* __restrict__ sur,
                                 const float* __restrict__ x,
                                 const float* __restrict__ mem,
                                 const float* __restrict__ s1,
                                 float* __restrict__ mem_new) {
  int wave = blockIdx.x * (blockDim.x >> 5) + (threadIdx.x >> 5);
  int mbase = (wave >> 3) * 16;
  int dbase = (wave & 7) * 16;
  v8f acc = {};
  for (int k = 0; k < NROW; k += 4) {
    v2f a = frag_cm(sur, MMEM, k, mbase);   // A[m][k] = sur[k][m]
    v2f b = frag_cm(x, DDIM, k, dbase);     // B[k][d] = x[k][d]
    acc = wmma4(a, b, acc);
  }
  int lane = threadIdx.x & 31;
  int dOff = lane & 15, mHi = (lane >> 4) * 8;
  const float c0 = 0.1f / (float)NROW;      // LR / (B*S)
#pragma unroll
  for (int i = 0; i < 8; ++i) {
    int m = mbase + mHi + i, d = dbase + dOff;
    float mm = mem[(size_t)m * DDIM + d];
    mem_new[(size_t)m * DDIM + d] = mm + c0 * (acc[i] - s1[m] * mm);
  }
}

// ------------- generic  out = [relu](src @ W^T + bias), K=128 --------------
__global__ void proj_kernel(const float* __restrict__ src,
                            const float* __restrict__ W,
                            const float* __restrict__ bias,
                            float* __restrict__ out,
                            int nrows, int ncols, int doRelu) {
  int wave = blockIdx.x * (blockDim.x >> 5) + (threadIdx.x >> 5);
  int ctiles = ncols >> 4;
  int rbase = (wave / ctiles) * 16;
  int cbase = (wave % ctiles) * 16;
  if (rbase >= nrows) return;              // wave-uniform guard
  v8f acc = {};
#pragma unroll
  for (int k = 0; k < DDIM; k += 4) {
    v2f a = frag_rm(src, DDIM, rbase, k);
    v2f b = frag_rm(W, DDIM, cbase, k);    // B[k][j] = W[j][k]
    acc = wmma4(a, b, acc);
  }
  int lane = threadIdx.x & 31;
  int cOff = lane & 15, rHi = (lane >> 4) * 8;
#pragma unroll
  for (int i = 0; i < 8; ++i) {
    int r = rbase + rHi + i, c = cbase + cOff;
    float v = acc[i] + bias[c];
    if (doRelu) v = fmaxf(v, 0.f);
    out[(size_t)r * ncols + c] = v;
  }
}

// ---------------- attention: flash-style online softmax --------------------
// wave = (b,h,sTile); 16 query rows per wave, stream M=512 keys in 16-chunks.
__global__ void attn_kernel(const float* __restrict__ qp,
                            const float* __restrict__ kp,
                            const float* __restrict__ vp,
                            float* __restrict__ ctx) {
  __shared__ float pls[8][16][18];         // per-wave P tile (C->A relayout)
  int wslot = threadIdx.x >> 5;
  int wave = blockIdx.x * (blockDim.x >> 5) + wslot;
  int b = wave >> 8, h = (wave >> 6) & 3, st = wave & 63;
  int qrow = b * 1024 + st * 16;
  int hd = h * 32;
  int lane = threadIdx.x & 31;
  int cOff = lane & 15, rHi = (lane >> 4) * 8;

  v8f acc0 = {}, acc1 = {};
  float rm[8], rs[8];
#pragma unroll
  for (int i = 0; i < 8; ++i) { rm[i] = -1e30f; rs[i] = 0.f; }
  const float scale = 0.17677669529663687f;   // 1/sqrt(hd=32)

  for (int mc = 0; mc < MMEM; mc += 16) {
    // scores tile 16x16, K = 32 (head slice)
    v8f sc = {};
#pragma unroll
    for (int k = 0; k < 32; k += 4) {
      v2f a = frag_rm(qp, DDIM, qrow, hd + k);
      v2f bf = frag_rm(kp, DDIM, mc, hd + k);   // B[k][m] = kp[m][hd+k]
      sc = wmma4(a, bf, sc);
    }
    float alpha[8];
#pragma unroll
    for (int i = 0; i < 8; ++i) {
      float v = sc[i] * scale;
      // row-max over the 16 lanes holding this row (xor 1,2,4,8 stays in half)
      float mx = v;
      mx = fmaxf(mx, __shfl_xor(mx, 1, 32));
      mx = fmaxf(mx, __shfl_xor(mx, 2, 32));
      mx = fmaxf(mx, __shfl_xor(mx, 4, 32));
      mx = fmaxf(mx, __shfl_xor(mx, 8, 32));
      float nm = fmaxf(rm[i], mx);
      alpha[i] = __expf(rm[i] - nm);
      float p = __expf(v - nm);
      float ps = p;
      ps += __shfl_xor(ps, 1, 32);
      ps += __shfl_xor(ps, 2, 32);
      ps += __shfl_xor(ps, 4, 32);
      ps += __shfl_xor(ps, 8, 32);
      rs[i] = rs[i] * alpha[i] + ps;
      rm[i] = nm;
      pls[wslot][rHi + i][cOff] = p;            // stage P for A-layout reload
      acc0[i] *= alpha[i];
      acc1[i] *= alpha[i];
    }
    // cross-lane LDS RAW hazard: wait for DS stores before lane-crossed reads
    asm volatile("s_wait_dscnt 0" ::: "memory");
    // ctx += P(16x16) @ V_chunk(16x32), K=16 in 4 wmma steps, 2 hd-column tiles
#pragma unroll
    for (int k = 0; k < 16; k += 4) {
      v2f a;
      int ak = k + ((lane >> 4) << 1);
      a.x = pls[wslot][lane & 15][ak];
      a.y = pls[wslot][lane & 15][ak + 1];
      v2f b0 = frag_cm(vp, DDIM, mc + k, hd);      // B[k][n] = vp[mc+k][hd+n]
      v2f b1 = frag_cm(vp, DDIM, mc + k, hd + 16);
      acc0 = wmma4(a, b0, acc0);
      acc1 = wmma4(a, b1, acc1);
    }
  }
#pragma unroll
  for (int i = 0; i < 8; ++i) {
    float inv = 1.f / rs[i];
    int r = qrow + rHi + i;
    ctx[(size_t)r * DDIM + hd + cOff]      = acc0[i] * inv;
    ctx[(size_t)r * DDIM + hd + 16 + cOff] = acc1[i] * inv;
  }
}

// ---------------------------------------------------------------------------
extern "C" void kernel_launch(void* const* d_in, const int* in_sizes, int n_in,
                              void* d_out, int out_size, void* d_ws, size_t ws_size,
                              hipStream_t stream) {
  const float* x      = (const float*)d_in[0];
  const float* memory = (const float*)d_in[1];
  const float* w_q    = (const float*)d_in[2];
  const float* w_k    = (const float*)d_in[3];
  const float* w_v    = (const float*)d_in[4];
  const float* b_q    = (const float*)d_in[5];
  const float* b_k    = (const float*)d_in[6];
  const float* b_v    = (const float*)d_in[7];
  const float* w_o    = (const float*)d_in[8];
  const float* b_o    = (const float*)d_in[9];
  const float* w_fc   = (const float*)d_in[10];
  const float* b_fc   = (const float*)d_in[11];
  float* out = (float*)d_out;

  float* ws      = (float*)d_ws;
  float* xn      = ws;                 // 2048
  float* mn      = ws + 2048;          // 512
  float* s1      = ws + 2560;          // 512
  float* mem_new = ws + 4096;          // 512*128   = 65536
  float* qp      = mem_new + 65536;    // 2048*128  = 262144
  float* kp      = qp + 262144;        // 512*128
  float* vp      = kp + 65536;         // 512*128
  float* ctx     = vp + 65536;         // 2048*128
  float* attn    = ctx + 262144;       // 2048*128
  float* sur     = attn + 262144;      // 2048*512  = 1048576  (~8.1 MB total)

  // 1) row norms (2048 + 512 rows, one wave each)
  norms_kernel<<<320, 256, 0, stream>>>(x, memory, xn, mn);
  // 2) surprise = sqrt(|x|^2 - 2 x.mem^T + |mem|^2)   [2048 x 512]
  surprise_kernel<<<512, 256, 0, stream>>>(x, memory, xn, mn, sur);
  // 3) column sums of surprise
  colsum_kernel<<<512, 256, 0, stream>>>(sur, s1);
  // 4) mem_new = mem + (LR/N)*(sur^T @ x - s1*mem)
  memupdate_kernel<<<32, 256, 0, stream>>>(sur, x, memory, s1, mem_new);
  // 5) projections q,k,v
  proj_kernel<<<128, 256, 0, stream>>>(x,       w_q, b_q, qp, NROW, DDIM, 0);
  proj_kernel<<<32,  256, 0, stream>>>(mem_new, w_k, b_k, kp, MMEM, DDIM, 0);
  proj_kernel<<<32,  256, 0, stream>>>(mem_new, w_v, b_v, vp, MMEM, DDIM, 0);
  // 6) attention (2*4*64 = 512 wave-tiles)
  attn_kernel<<<64, 256, 0, stream>>>(qp, kp, vp, ctx);
  // 7) output projection
  proj_kernel<<<128, 256, 0, stream>>>(ctx, w_o, b_o, attn, NROW, DDIM, 0);
  // 8) FC + ReLU -> d_out [2048 x 512]
  proj_kernel<<<512, 256, 0, stream>>>(attn, w_fc, b_fc, out, NROW, HOUT, 1);
}